// GCN_13881334300836
// MI455X (gfx1250) — compile-verified
//
#include <hip/hip_runtime.h>

typedef __attribute__((ext_vector_type(2))) float v2f;
typedef __attribute__((ext_vector_type(8))) float v8f;

#define LDS_STRIDE 68   // 64 + 4: conflict-free banks, keeps float4 alignment

// ---------------- degree / norm kernels ----------------

__global__ void init_deg_kernel(float* __restrict__ deg, int n) {
  int i = blockIdx.x * blockDim.x + threadIdx.x;
  if (i < n) deg[i] = 1.0f;  // self-loop contributes 1 to in-degree
}

__global__ void deg_edges_kernel(const int* __restrict__ dst, float* __restrict__ deg, int e) {
  int i = blockIdx.x * blockDim.x + threadIdx.x;
  if (i < e) atomicAdd(&deg[dst[i]], 1.0f);
}

__global__ void rsqrt_kernel(float* __restrict__ d, int n) {
  int i = blockIdx.x * blockDim.x + threadIdx.x;
  if (i < n) d[i] = rsqrtf(d[i]);  // deg >= 1 always (self-loops)
}

// ---------------- WMMA GEMM: T[N,64] = A[N,64] @ W[64,64] ----------------
// 256 threads = 8 waves; each wave computes a 16-row stripe x 64 cols.

__global__ __launch_bounds__(256)
void gemm_wmma_kernel(const float* __restrict__ A, const float* __restrict__ W,
                      float* __restrict__ T, int n) {
  __shared__ float ldsW[64 * LDS_STRIDE];       // 17.4 KB
  __shared__ float ldsA[8 * 16 * LDS_STRIDE];   // 34.8 KB

  const int tid  = threadIdx.x;
  const int wave = tid >> 5;
  const int lane = tid & 31;
  const int half = lane >> 4;   // 0: K+0/K+1, 1: K+2/K+3 (32-bit A/B layout)
  const int l16  = lane & 15;
  const int r0   = blockIdx.x * 128 + wave * 16;

  // Stage W (64x64) into LDS, float4-coalesced across the block.
  for (int i = tid; i < 1024; i += 256) {
    int row = i >> 4;
    int col = (i & 15) << 2;
    float4 w4 = *reinterpret_cast<const float4*>(W + row * 64 + col);
    *reinterpret_cast<float4*>(&ldsW[row * LDS_STRIDE + col]) = w4;
  }

  // Stage this wave's 16x64 A tile into LDS (zeros past tail).
  float* tile = &ldsA[wave * 16 * LDS_STRIDE];
#pragma unroll
  for (int i = 0; i < 8; ++i) {
    int f4  = i * 32 + lane;       // 256 float4s per tile
    int row = f4 >> 4;
    int col = (f4 & 15) << 2;
    float4 v = make_float4(0.f, 0.f, 0.f, 0.f);
    int r = r0 + row;
    if (r < n) v = *reinterpret_cast<const float4*>(A + (size_t)r * 64 + col);
    *reinterpret_cast<float4*>(&tile[row * LDS_STRIDE + col]) = v;
  }
  __syncthreads();

  v8f acc0 = {}, acc1 = {}, acc2 = {}, acc3 = {};
  const float* arow = &tile[l16 * LDS_STRIDE];

#pragma unroll
  for (int kk = 0; kk < 64; kk += 4) {
    int ka = kk + half * 2;
    v2f a;
    a.x = arow[ka];
    a.y = arow[ka + 1];
    const float* w0 = &ldsW[ka * LDS_STRIDE];
    const float* w1 = &ldsW[(ka + 1) * LDS_STRIDE];
    v2f b0, b1, b2, b3;
    b0.x = w0[l16];      b0.y = w1[l16];
    b1.x = w0[16 + l16]; b1.y = w1[16 + l16];
    b2.x = w0[32 + l16]; b2.y = w1[32 + l16];
    b3.x = w0[48 + l16]; b3.y = w1[48 + l16];
    acc0 = __builtin_amdgcn_wmma_f32_16x16x4_f32(false, a, false, b0, (short)0, acc0, false, false);
    acc1 = __builtin_amdgcn_wmma_f32_16x16x4_f32(false, a, false, b1, (short)0, acc1, false, false);
    acc2 = __builtin_amdgcn_wmma_f32_16x16x4_f32(false, a, false, b2, (short)0, acc2, false, false);
    acc3 = __builtin_amdgcn_wmma_f32_16x16x4_f32(false, a, false, b3, (short)0, acc3, false, false);
  }

  // D layout: VGPR r -> row (half*8 + r), col = l16 within each 16-col tile.
  int mbase = r0 + half * 8;
#pragma unroll
  for (int r = 0; r < 8; ++r) {
    int m = mbase + r;
    if (m < n) {
      float* o = T + (size_t)m * 64 + l16;
      o[0]  = acc0[r];
      o[16] = acc1[r];
      o[32] = acc2[r];
      o[48] = acc3[r];
    }
  }
}

// ---------------- aggregation ----------------

// agg[i,:] = dinv[i]^2 * T[i,:]   (self-loop term; also zero-inits accumulator)
__global__ void init_agg_kernel(const float* __restrict__ T, const float* __restrict__ dinv,
                                float* __restrict__ agg, int n64) {
  int idx = blockIdx.x * blockDim.x + threadIdx.x;
  if (idx < n64) {
    float d = dinv[idx >> 6];
    agg[idx] = d * d * T[idx];
  }
}

// one wave32 per edge: lane f handles features f and f+32 (coalesced 128B bursts)
__global__ void scatter_kernel(const float* __restrict__ T, const int* __restrict__ src,
                               const int* __restrict__ dst, const float* __restrict__ dinv,
                               float* __restrict__ agg, int e) {
  int gid  = blockIdx.x * blockDim.x + threadIdx.x;
  int edge = gid >> 5;
  int lane = gid & 31;
  if (edge >= e) return;
  int s = src[edge];
  int d = dst[edge];
  float w = dinv[s] * dinv[d];
  const float* ts = T + (size_t)s * 64;
  float* ad = agg + (size_t)d * 64;
  atomicAdd(&ad[lane],      w * ts[lane]);
  atomicAdd(&ad[lane + 32], w * ts[lane + 32]);
}

__global__ void bias_act_kernel(float* __restrict__ h, const float* __restrict__ b,
                                int n64, int relu) {
  int idx = blockIdx.x * blockDim.x + threadIdx.x;
  if (idx < n64) {
    float v = h[idx] + b[idx & 63];
    h[idx] = relu ? fmaxf(v, 0.0f) : v;
  }
}

// ---------------- pooling + classifier ----------------

__global__ void pool_init_kernel(float* __restrict__ pooled, float* __restrict__ counts, int g) {
  int idx = blockIdx.x * blockDim.x + threadIdx.x;
  if (idx < g * 64) pooled[idx] = 0.0f;
  if (idx < g) counts[idx] = 0.0f;
}

__global__ void pool_kernel(const float* __restrict__ h, const int* __restrict__ batch,
                            float* __restrict__ pooled, float* __restrict__ counts, int n64) {
  int idx = blockIdx.x * blockDim.x + threadIdx.x;
  if (idx < n64) {
    int i = idx >> 6;
    int f = idx & 63;
    int g = batch[i];
    atomicAdd(&pooled[g * 64 + f], h[idx]);
    if (f == 0) atomicAdd(&counts[g], 1.0f);
  }
}

__global__ void final_kernel(const float* __restrict__ pooled, const float* __restrict__ counts,
                             const float* __restrict__ Wl, const float* __restrict__ bl,
                             float* __restrict__ out, int g, int c) {
  int idx = blockIdx.x * blockDim.x + threadIdx.x;
  if (idx < g * c) {
    int gi = idx / c;
    int ci = idx - gi * c;
    const float* p = pooled + gi * 64;
    float s = 0.0f;
#pragma unroll
    for (int k = 0; k < 64; ++k) s += p[k] * Wl[k * c + ci];
    out[idx] = s / fmaxf(counts[gi], 1.0f) + bl[ci];
  }
}

// ---------------- host side ----------------

extern "C" void kernel_launch(void* const* d_in, const int* in_sizes, int n_in,
                              void* d_out, int out_size, void* d_ws, size_t ws_size,
                              hipStream_t stream) {
  (void)n_in; (void)ws_size;
  const float* x    = (const float*)d_in[0];
  const int*   ei   = (const int*)d_in[1];
  const int*   bat  = (const int*)d_in[2];
  const float* W1   = (const float*)d_in[3];
  const float* b1   = (const float*)d_in[4];
  const float* W2   = (const float*)d_in[5];
  const float* b2   = (const float*)d_in[6];
  const float* W3   = (const float*)d_in[7];
  const float* b3   = (const float*)d_in[8];
  const float* Wl   = (const float*)d_in[9];
  const float* bl   = (const float*)d_in[10];

  const int N = in_sizes[0] / 64;
  const int E = in_sizes[1] / 2;
  const int C = 10;
  const int G = out_size / C;
  const int* src = ei;
  const int* dst = ei + E;

  float* ws     = (float*)d_ws;
  float* dinv   = ws;
  float* T      = dinv + ((N + 255) & ~255);
  float* Hb     = T + (size_t)N * 64;
  float* pooled = Hb + (size_t)N * 64;
  float* counts = pooled + (size_t)G * 64;

  const int B   = 256;
  const int n64 = N * 64;

  // symmetric normalization: deg (incl. self-loop) -> dinv = rsqrt(deg)
  init_deg_kernel<<<(N + B - 1) / B, B, 0, stream>>>(dinv, N);
  deg_edges_kernel<<<(E + B - 1) / B, B, 0, stream>>>(dst, dinv, E);
  rsqrt_kernel<<<(N + B - 1) / B, B, 0, stream>>>(dinv, N);

  const float* Ws[3] = {W1, W2, W3};
  const float* bs[3] = {b1, b2, b3};
  const float* inl = x;
  for (int l = 0; l < 3; ++l) {
    gemm_wmma_kernel<<<(N + 127) / 128, 256, 0, stream>>>(inl, Ws[l], T, N);
    init_agg_kernel<<<(n64 + B - 1) / B, B, 0, stream>>>(T, dinv, Hb, n64);
    scatter_kernel<<<(E * 32 + B - 1) / B, B, 0, stream>>>(T, src, dst, dinv, Hb, E);
    bias_act_kernel<<<(n64 + B - 1) / B, B, 0, stream>>>(Hb, bs[l], n64, l < 2 ? 1 : 0);
    inl = Hb;
  }

  pool_init_kernel<<<(G * 64 + B - 1) / B, B, 0, stream>>>(pooled, counts, G);
  pool_kernel<<<(n64 + B - 1) / B, B, 0, stream>>>(Hb, bat, pooled, counts, n64);
  final_kernel<<<(G * C + B - 1) / B, B, 0, stream>>>(pooled, counts, Wl, bl, (float*)d_out, G, C);
}